// MoNet_1211180777632
// MI455X (gfx1250) — compile-verified
//
#include <hip/hip_runtime.h>
#include <hip/hip_bf16.h>

// ---------------------------------------------------------------------------
// MoNet (GMMConv x2) for MI455X / gfx1250, wave32.
// GEMMs use V_WMMA_F32_16X16X4_F32 (fp32-exact matrix core path).
// Edge phase: folded-K gather + hardware f32 global atomics (L2-resident).
// ---------------------------------------------------------------------------

typedef __attribute__((ext_vector_type(2))) float v2f;
typedef __attribute__((ext_vector_type(8))) float v8f;

#define N_NODES 50000
#define N_EDGES 800000
#define IN_F    128
#define HID     64
#define OUT_F   40
#define KK      3

// ---------------------------------------------------------------------------
// Edge Gaussian kernel weights: w[e,k] = exp(-0.5*||(tanh(W p + b) - mu_k) * isig_k||^2)
// ---------------------------------------------------------------------------
__global__ void edge_w_kernel(const float* __restrict__ pseudo,
                              const float* __restrict__ ppw,   // (2,2) row-major
                              const float* __restrict__ ppb,   // (2,)
                              const float* __restrict__ mu,    // (3,2)
                              const float* __restrict__ isig,  // (3,2)
                              float* __restrict__ w, int E)
{
    int e = blockIdx.x * blockDim.x + threadIdx.x;
    if (e >= E) return;
    v2f ps = *(const v2f*)(pseudo + 2 * (size_t)e);
    float p0 = tanhf(ppw[0] * ps.x + ppw[1] * ps.y + ppb[0]);
    float p1 = tanhf(ppw[2] * ps.x + ppw[3] * ps.y + ppb[1]);
#pragma unroll
    for (int k = 0; k < KK; ++k) {
        float dx = (p0 - mu[2 * k + 0]) * isig[2 * k + 0];
        float dy = (p1 - mu[2 * k + 1]) * isig[2 * k + 1];
        w[3 * (size_t)e + k] = __expf(-0.5f * (dx * dx + dy * dy));
    }
}

// ---------------------------------------------------------------------------
// Generic fp32 WMMA GEMM:  D(M x n) = A(M x K) @ Bt(n x K)^T
// One wave computes a 16x32 strip of D (two 16x16 accumulators share A).
// A 16x4 fp32 operand layout: lanes 0-15 row M=m0+l, float2 = K{k,k+1};
// lanes 16-31 same rows, K{k+2,k+3}. B(4x16) symmetric. D: VGPR j ->
// row m0+j (lanes 0-15) / m0+8+j (lanes 16-31), col = n0 + (lane&15).
// ---------------------------------------------------------------------------
__global__ void gemm_wmma_f32(const float* __restrict__ A, int lda,
                              const float* __restrict__ Bt, int ldb,
                              float* __restrict__ D, int ldd, int n_valid,
                              int n_groups, int total_waves, int Kdim)
{
    int wave = (blockIdx.x * blockDim.x + threadIdx.x) >> 5;
    int lane = threadIdx.x & 31;
    if (wave >= total_waves) return;              // wave-uniform: EXEC stays all-1s

    int tm = wave / n_groups, tn = wave % n_groups;
    int m0 = tm << 4, n0 = tn << 5;
    int l15 = lane & 15;
    int kh  = (lane >> 4) << 1;                   // K sub-offset 0 or 2

    const float* ap = A + (size_t)(m0 + l15) * lda + kh;

    int nA = n0 + l15, nB = nA + 16;
    float mAv = (nA < n_valid) ? 1.0f : 0.0f;
    float mBv = (nB < n_valid) ? 1.0f : 0.0f;
    const float* bpA = Bt + (size_t)((nA < n_valid) ? nA : 0) * ldb + kh;
    const float* bpB = Bt + (size_t)((nB < n_valid) ? nB : 0) * ldb + kh;

    v8f c0 = {};
    v8f c1 = {};
    for (int k = 0; k < Kdim; k += 4) {
        v2f a  = *(const v2f*)(ap  + k);
        v2f b0 = *(const v2f*)(bpA + k);
        v2f b1 = *(const v2f*)(bpB + k);
        b0 *= mAv;
        b1 *= mBv;
        c0 = __builtin_amdgcn_wmma_f32_16x16x4_f32(false, a, false, b0, (short)0, c0, false, false);
        c1 = __builtin_amdgcn_wmma_f32_16x16x4_f32(false, a, false, b1, (short)0, c1, false, false);
    }

    int rbase = m0 + ((lane >> 4) << 3);
#pragma unroll
    for (int j = 0; j < 8; ++j) {
        size_t roff = (size_t)(rbase + j) * ldd;
        if (nA < n_valid) D[roff + nA] = c0[j];
        if (nB < n_valid) D[roff + nB] = c1[j];
    }
}

// ---------------------------------------------------------------------------
// Init a (rows x ncols) buffer to a broadcast bias row (also zeroes atomics' base).
// ---------------------------------------------------------------------------
__global__ void init_bias_kernel(float* __restrict__ dst, const float* __restrict__ bias,
                                 int ncols, int total)
{
    int i = blockIdx.x * blockDim.x + threadIdx.x;
    if (i < total) dst[i] = bias[i % ncols];
}

// ---------------------------------------------------------------------------
// Layer-0 scatter: per (edge, o<64): v = sum_k w[e,k]*hp[src,k*64+o]; h[dst,o] += v
// ---------------------------------------------------------------------------
__global__ void scatter0_kernel(const float* __restrict__ hp, const float* __restrict__ w,
                                const int* __restrict__ src, const int* __restrict__ dst,
                                float* __restrict__ h, int E)
{
    int i = blockIdx.x * blockDim.x + threadIdx.x;
    if (i >= E * 64) return;
    int e = i >> 6;
    int o = i & 63;
    int s = src[e], d = dst[e];
    const float* hs = hp + (size_t)s * (KK * HID);
    float w0 = w[3 * (size_t)e + 0], w1 = w[3 * (size_t)e + 1], w2 = w[3 * (size_t)e + 2];
    float v = w0 * hs[o] + w1 * hs[HID + o] + w2 * hs[2 * HID + o];
    unsafeAtomicAdd(&h[(size_t)d * HID + o], v);
}

// ---------------------------------------------------------------------------
// Layer-1 scatter: per (edge, o<40): v = sum_k w[e,k]*hp1[src,k*40+o]; out[dst,o] += v
// hp1 row stride padded to 128.
// ---------------------------------------------------------------------------
__global__ void scatter1_kernel(const float* __restrict__ hp, const float* __restrict__ w,
                                const int* __restrict__ src, const int* __restrict__ dst,
                                float* __restrict__ out, int E)
{
    int i = blockIdx.x * blockDim.x + threadIdx.x;
    if (i >= E * OUT_F) return;
    int e = i / OUT_F;
    int o = i - e * OUT_F;
    int s = src[e], d = dst[e];
    const float* hs = hp + (size_t)s * 128;
    float w0 = w[3 * (size_t)e + 0], w1 = w[3 * (size_t)e + 1], w2 = w[3 * (size_t)e + 2];
    float v = w0 * hs[o] + w1 * hs[OUT_F + o] + w2 * hs[2 * OUT_F + o];
    unsafeAtomicAdd(&out[(size_t)d * OUT_F + o], v);
}

// ---------------------------------------------------------------------------
extern "C" void kernel_launch(void* const* d_in, const int* in_sizes, int n_in,
                              void* d_out, int out_size, void* d_ws, size_t ws_size,
                              hipStream_t stream)
{
    const float* feat   = (const float*)d_in[0];
    const float* pseudo = (const float*)d_in[1];
    const int*   src    = (const int*)d_in[2];
    const int*   dst    = (const int*)d_in[3];
    const float* pp0_w  = (const float*)d_in[4];
    const float* pp0_b  = (const float*)d_in[5];
    const float* fc0    = (const float*)d_in[6];
    const float* mu0    = (const float*)d_in[7];
    const float* is0    = (const float*)d_in[8];
    const float* b0     = (const float*)d_in[9];
    const float* pp1_w  = (const float*)d_in[10];
    const float* pp1_b  = (const float*)d_in[11];
    const float* fc1    = (const float*)d_in[12];
    const float* mu1    = (const float*)d_in[13];
    const float* is1    = (const float*)d_in[14];
    const float* b1     = (const float*)d_in[15];
    float* out = (float*)d_out;

    // Workspace layout (buffer reuse; lifetimes are stream-ordered):
    //   [0,        9.6 MB)  w   : edge weights (E x 3), reused for both layers
    //   [9.6 MB,  48.0 MB)  hp  : hp0 (N x 192), then hp1 (N x 128, 120 valid)
    //   [48 MB,   60.8 MB)  h   : layer-0 node output (N x 64), init to b0
    char*  ws = (char*)d_ws;
    float* w  = (float*)(ws);
    float* hp = (float*)(ws + 9600000);
    float* h  = (float*)(ws + 48000000);

    const int E = N_EDGES;

    // ---- Layer 0 -----------------------------------------------------------
    edge_w_kernel<<<(E + 255) / 256, 256, 0, stream>>>(pseudo, pp0_w, pp0_b, mu0, is0, w, E);

    // hp0 = feat @ fc0.T : (50000,128)x(128,192); 3125 m-tiles, 6 n-groups
    {
        int m_tiles = N_NODES / 16, n_groups = (KK * HID) / 32;
        int total_waves = m_tiles * n_groups;
        int blocks = (total_waves * 32 + 255) / 256;
        gemm_wmma_f32<<<blocks, 256, 0, stream>>>(feat, IN_F, fc0, IN_F,
                                                  hp, KK * HID, KK * HID,
                                                  n_groups, total_waves, IN_F);
    }

    init_bias_kernel<<<(N_NODES * HID + 255) / 256, 256, 0, stream>>>(h, b0, HID, N_NODES * HID);
    scatter0_kernel<<<(E * 64 + 255) / 256, 256, 0, stream>>>(hp, w, src, dst, h, E);

    // ---- Layer 1 -----------------------------------------------------------
    edge_w_kernel<<<(E + 255) / 256, 256, 0, stream>>>(pseudo, pp1_w, pp1_b, mu1, is1, w, E);

    // hp1 = h @ fc1.T : (50000,64)x(64,120); stored with row stride 128, cols 120 valid
    {
        int m_tiles = N_NODES / 16, n_groups = 128 / 32;
        int total_waves = m_tiles * n_groups;
        int blocks = (total_waves * 32 + 255) / 256;
        gemm_wmma_f32<<<blocks, 256, 0, stream>>>(h, HID, fc1, HID,
                                                  hp, 128, KK * OUT_F,
                                                  n_groups, total_waves, HID);
    }

    init_bias_kernel<<<(N_NODES * OUT_F + 255) / 256, 256, 0, stream>>>(out, b1, OUT_F,
                                                                        N_NODES * OUT_F);
    scatter1_kernel<<<(E * OUT_F + 255) / 256, 256, 0, stream>>>(hp, w, src, dst, out, E);
}